// my_rnn_46712064311515
// MI455X (gfx1250) — compile-verified
//
#include <hip/hip_runtime.h>
#include <math.h>

// ---------------------------------------------------------------------------
// Bidirectional 2-layer tanh-RNN for MI455X (gfx1250, wave32, WMMA).
//   B=64, N=2048, E=256, H=256
// Strategy:
//   * 4 input-projection GEMMs with V_WMMA_F32_16X16X4_F32, LDS-tiled.
//   * 4 sequential scans; each direction = one 1024-thread workgroup that
//     keeps Whh (256KB) + h (64KB) resident in the full 320KB WGP LDS and
//     does the per-step (64x256)@(256x256)^T with f32 WMMA.
//   * Branch-free tanh (native v_tanh_f32 when available) and software-
//     pipelined xproj loads keep EXEC uniform and hide HBM latency behind
//     the WMMA phase of the next timestep.
// ---------------------------------------------------------------------------

typedef float v2f __attribute__((ext_vector_type(2)));
typedef float v8f __attribute__((ext_vector_type(8)));

#define BATCH 64
#define SEQ   2048
#define EDIM  256
#define HDIM  256
#define MROWS (BATCH * SEQ)

__device__ __forceinline__ v8f wmma_f32(v2f a, v2f b, v8f c) {
  // D = A(16x4) x B(4x16) + C(16x16), all f32, wave32.
  return __builtin_amdgcn_wmma_f32_16x16x4_f32(
      /*neg_a=*/false, a, /*neg_b=*/false, b,
      /*c_mod=*/(short)0, c, /*reuse_a=*/false, /*reuse_b=*/false);
}

// Branch-free tanh: keeps EXEC all-1s (no saveexec region on the scan's
// critical path). gfx1250 has native V_TANH_F32; fall back to
// 1 - 2/(exp2(2*log2e*x) + 1), which saturates correctly at +-inf.
__device__ __forceinline__ float fast_tanh(float x) {
#if __has_builtin(__builtin_amdgcn_tanhf)
  return __builtin_amdgcn_tanhf(x);
#else
  const float e = __builtin_amdgcn_exp2f(x * 2.885390081777927f);  // 2*log2(e)
  return 1.0f - 2.0f * __builtin_amdgcn_rcpf(e + 1.0f);
#endif
}

// ---------------------------------------------------------------------------
// GEMM + bias: out(M,256) = A(M,K) @ W(256,K)^T + bias
// block = 128 threads (4 waves), tile = 64(M) x 64(N), K chunked by 64 in LDS.
// Wave w computes N columns [w*16, w*16+16) for all 64 M rows: per k-step one
// B fragment feeds 4 WMMAs.
// ---------------------------------------------------------------------------
template <int K>
__global__ __launch_bounds__(128) void gemm_bias(const float* __restrict__ A,
                                                 const float* __restrict__ W,
                                                 const float* __restrict__ bias,
                                                 float* __restrict__ out) {
  __shared__ float Asm[64 * 64];
  __shared__ float Bsm[64 * 64];

  const int tid  = threadIdx.x;
  const int lane = tid & 31;
  const int wave = tid >> 5;
  const int l15  = lane & 15;
  const int hi   = lane >> 4;       // 0 | 1
  const int klo  = hi << 1;         // lane-half k offset (0 or 2)

  const int m0 = blockIdx.y * 64;
  const int n0 = blockIdx.x * 64;

  v8f zero = {};
  v8f acc[4];
#pragma unroll
  for (int i = 0; i < 4; ++i) acc[i] = zero;

  for (int kc = 0; kc < K; kc += 64) {
    // Cooperative 16B-vector copies: 64 rows x 16 float4 per matrix.
#pragma unroll
    for (int i = 0; i < 8; ++i) {
      const int idx = tid + i * 128;
      const int row = idx >> 4;
      const int c4  = (idx & 15) << 2;
      *(float4*)&Asm[row * 64 + c4] =
          *(const float4*)&A[(size_t)(m0 + row) * K + kc + c4];
      *(float4*)&Bsm[row * 64 + c4] =
          *(const float4*)&W[(size_t)(n0 + row) * K + kc + c4];
      // Prefetch the next K chunk while this one is consumed.
      if (kc + 64 < K) {
        __builtin_prefetch(&A[(size_t)(m0 + row) * K + kc + 64 + c4], 0, 1);
        __builtin_prefetch(&W[(size_t)(n0 + row) * K + kc + 64 + c4], 0, 1);
      }
    }
    __syncthreads();

    const int nsub = wave * 16;
#pragma unroll 4
    for (int kk = 0; kk < 64; kk += 4) {
      const int ko = kk + klo;
      // B fragment (4x16): lane<16 holds K=+0,+1; lane>=16 holds K=+2,+3.
      v2f bf = *(const v2f*)&Bsm[(nsub + l15) * 64 + ko];
#pragma unroll
      for (int mt = 0; mt < 4; ++mt) {
        v2f af = *(const v2f*)&Asm[(mt * 16 + l15) * 64 + ko];
        acc[mt] = wmma_f32(af, bf, acc[mt]);
      }
    }
    __syncthreads();
  }

  const int ncol = n0 + wave * 16 + l15;
  const float bv = bias[ncol];
#pragma unroll
  for (int mt = 0; mt < 4; ++mt) {
#pragma unroll
    for (int r = 0; r < 8; ++r) {
      const int m = m0 + mt * 16 + r + (hi << 3);  // C layout: vgpr r -> M=r(+8)
      out[(size_t)m * HDIM + ncol] = acc[mt][r] + bv;
    }
  }
}

// ---------------------------------------------------------------------------
// One bidirectional scan layer: grid = {fwd, bwd}, 1024 threads (32 waves).
// LDS: Whh 256KB + h 64KB = full 320KB WGP LDS.
// Per step: each wave computes 2 of the 64 (16x16) output tiles with 128
// f32 WMMAs, adds xproj+bias, tanh, republishes h to LDS. xproj for step
// t+1 is loaded right after step t's values are consumed, so its latency
// hides behind the next step's WMMA phase.
// ---------------------------------------------------------------------------
__global__ __launch_bounds__(1024) void birnn_scan(
    const float* __restrict__ xproj_f, const float* __restrict__ xproj_b,
    const float* __restrict__ Whh_f, const float* __restrict__ bhh_f,
    const float* __restrict__ Whh_b, const float* __restrict__ bhh_b,
    float* __restrict__ seq_out,    // (BATCH, SEQ, 2H) or nullptr
    float* __restrict__ final_out)  // (BATCH, 2H) or nullptr
{
  extern __shared__ float smem[];
  float* Wlds = smem;                 // 256*256 f32 = 256KB, stored [n][k]
  float* hlds = smem + HDIM * HDIM;   // 64*256  f32 =  64KB, stored [b][k]

  const int rev = blockIdx.x;  // 0 = forward, 1 = backward (reverse scan)
  const float* __restrict__ xproj = rev ? xproj_b : xproj_f;
  const float* __restrict__ Whh   = rev ? Whh_b : Whh_f;
  const float* __restrict__ bhh   = rev ? bhh_b : bhh_f;
  const int col_off = rev ? HDIM : 0;

  const int tid  = threadIdx.x;
  const int lane = tid & 31;
  const int wave = tid >> 5;
  const int l15  = lane & 15;
  const int hi   = lane >> 4;
  const int klo  = hi << 1;

  // Stage Whh into LDS once (resident for all 2048 steps), zero h.
#pragma unroll
  for (int i = 0; i < 16; ++i)
    ((float4*)Wlds)[tid + i * 1024] = ((const float4*)Whh)[tid + i * 1024];
#pragma unroll
  for (int i = 0; i < 16; ++i) hlds[tid + i * 1024] = 0.0f;
  __syncthreads();

  // Wave -> tile mapping: 64 tiles (4 M-tiles x 16 N-tiles), 2 tiles/wave
  // sharing the A fragment.
  const int mt  = wave & 3;    // batch tile: rows mt*16 .. mt*16+15
  const int np  = wave >> 2;   // column pair 0..7
  const int n_a = np * 32 + l15;
  const int n_b = n_a + 16;
  const float bias_a = bhh[n_a];
  const float bias_b = bhh[n_b];

  // Software pipeline: xa/xb hold xproj values for the step being computed.
  float xa[8], xb[8];
  {
    const int t0 = rev ? (SEQ - 1) : 0;
#pragma unroll
    for (int r = 0; r < 8; ++r) {
      const int b = mt * 16 + r + (hi << 3);
      const float* xp = xproj + ((size_t)b * SEQ + t0) * HDIM;
      xa[r] = xp[n_a];
      xb[r] = xp[n_b];
    }
  }

  const v8f zero = {};
  for (int t = 0; t < SEQ; ++t) {
    const int tt = rev ? (SEQ - 1 - t) : t;

    v8f acc0 = zero, acc1 = zero;
#pragma unroll 4
    for (int kk = 0; kk < HDIM; kk += 4) {
      const int ko = kk + klo;
      v2f af = *(const v2f*)&hlds[(mt * 16 + l15) * HDIM + ko];
      v2f b0 = *(const v2f*)&Wlds[n_a * HDIM + ko];
      v2f b1 = *(const v2f*)&Wlds[n_b * HDIM + ko];
      acc0 = wmma_f32(af, b0, acc0);
      acc1 = wmma_f32(af, b1, acc1);
    }

    // h_new = tanh(hh + xproj_t + bhh)   (branch-free, EXEC stays all-1s)
    float h0[8], h1[8];
#pragma unroll
    for (int r = 0; r < 8; ++r) {
      h0[r] = fast_tanh(acc0[r] + xa[r] + bias_a);
      h1[r] = fast_tanh(acc1[r] + xb[r] + bias_b);
    }

    // Issue next timestep's xproj loads now; first use is after the next
    // step's 128-WMMA phase, so the memory latency is fully hidden.
    if (t + 1 < SEQ) {
      const int tn = rev ? (tt - 1) : (tt + 1);
#pragma unroll
      for (int r = 0; r < 8; ++r) {
        const int b = mt * 16 + r + (hi << 3);
        const float* xp = xproj + ((size_t)b * SEQ + tn) * HDIM;
        xa[r] = xp[n_a];
        xb[r] = xp[n_b];
      }
    }

    __syncthreads();  // all waves done reading h for this step
#pragma unroll
    for (int r = 0; r < 8; ++r) {
      const int b = mt * 16 + r + (hi << 3);
      hlds[b * HDIM + n_a] = h0[r];
      hlds[b * HDIM + n_b] = h1[r];
      if (seq_out) {
        float* so = seq_out + ((size_t)b * SEQ + tt) * (2 * HDIM) + col_off;
        so[n_a] = h0[r];
        so[n_b] = h1[r];
      }
    }
    __syncthreads();  // h republished before next step's reads
  }

  if (final_out) {
#pragma unroll
    for (int r = 0; r < 8; ++r) {
      const int b = mt * 16 + r + (hi << 3);
      final_out[(size_t)b * (2 * HDIM) + col_off + n_a] = hlds[b * HDIM + n_a];
      final_out[(size_t)b * (2 * HDIM) + col_off + n_b] = hlds[b * HDIM + n_b];
    }
  }
}

// ---------------------------------------------------------------------------
extern "C" void kernel_launch(void* const* d_in, const int* in_sizes, int n_in,
                              void* d_out, int out_size, void* d_ws,
                              size_t ws_size, hipStream_t stream) {
  const float* x      = (const float*)d_in[0];
  const float* Wih_f0 = (const float*)d_in[1];
  const float* bih_f0 = (const float*)d_in[2];
  const float* Whh_f0 = (const float*)d_in[3];
  const float* bhh_f0 = (const float*)d_in[4];
  const float* Wih_b0 = (const float*)d_in[5];
  const float* bih_b0 = (const float*)d_in[6];
  const float* Whh_b0 = (const float*)d_in[7];
  const float* bhh_b0 = (const float*)d_in[8];
  const float* Wih_f1 = (const float*)d_in[9];
  const float* bih_f1 = (const float*)d_in[10];
  const float* Whh_f1 = (const float*)d_in[11];
  const float* bhh_f1 = (const float*)d_in[12];
  const float* Wih_b1 = (const float*)d_in[13];
  const float* bih_b1 = (const float*)d_in[14];
  const float* Whh_b1 = (const float*)d_in[15];
  const float* bhh_b1 = (const float*)d_in[16];

  float* ws = (float*)d_ws;
  const size_t MH = (size_t)MROWS * HDIM;   // 33.5M floats
  float* xpA  = ws;                         // (M, 256)
  float* xpB  = ws + MH;                    // (M, 256)
  float* out0 = ws + 2 * MH;                // (M, 512) = layer-0 concat output

  const dim3 ggrid(HDIM / 64, MROWS / 64);  // (4, 2048)
  const size_t scan_lds = (size_t)(HDIM * HDIM + BATCH * HDIM) * sizeof(float);

  // Layer 0 input projections, then bidirectional scan writing out0.
  gemm_bias<EDIM><<<ggrid, 128, 0, stream>>>(x, Wih_f0, bih_f0, xpA);
  gemm_bias<EDIM><<<ggrid, 128, 0, stream>>>(x, Wih_b0, bih_b0, xpB);
  birnn_scan<<<2, 1024, scan_lds, stream>>>(xpA, xpB, Whh_f0, bhh_f0, Whh_b0,
                                            bhh_b0, out0, nullptr);

  // Layer 1 input projections (K = 512), scan emits only final hidden states.
  gemm_bias<2 * HDIM><<<ggrid, 128, 0, stream>>>(out0, Wih_f1, bih_f1, xpA);
  gemm_bias<2 * HDIM><<<ggrid, 128, 0, stream>>>(out0, Wih_b1, bih_b1, xpB);
  birnn_scan<<<2, 1024, scan_lds, stream>>>(xpA, xpB, Whh_f1, bhh_f1, Whh_b1,
                                            bhh_b1, nullptr, (float*)d_out);
}